// FlowGNN_LP_59442347377025
// MI455X (gfx1250) — compile-verified
//
#include <hip/hip_runtime.h>
#include <math.h>

#define NN   20000
#define DD   8
#define INF  64
#define HIDF 128
#define OUTF 64
#define EE   100000
#define MM   (NN * (DD + 1))   /* 180000 */
#define EPSF 1e-5f

typedef __attribute__((ext_vector_type(2))) float v2f;
typedef __attribute__((ext_vector_type(8))) float v8f;

// ---------------------------------------------------------------------------
// init: zero batch-stat accumulators + loss output
// ---------------------------------------------------------------------------
__global__ void k_init(float* __restrict__ stats, float* __restrict__ out) {
  int i = blockIdx.x * blockDim.x + threadIdx.x;
  if (i < 768) stats[i] = 0.f;
  if (i == 0) out[0] = 0.f;
}

// ---------------------------------------------------------------------------
// Fused: level-1 temporal gather + masked mean (into LDS) -> agg1 tile,
// then z1 = agg1 @ W1 + b1 via V_WMMA_F32_16X16X4_F32, with fused
// per-column batch statistics (sum / sumsq) for BatchNorm1.
// One block = 16 rows of M x all 128 columns (8 waves x 16 cols each).
// ---------------------------------------------------------------------------
__global__ __launch_bounds__(256)
void k_gemm1(const float* __restrict__ x,
             const float* __restrict__ node_ts,
             const int*   __restrict__ nidx,
             const float* __restrict__ nts,
             const float* __restrict__ W1,
             const float* __restrict__ b1,
             float* __restrict__ z1,
             float* __restrict__ sum1,
             float* __restrict__ sumsq1) {
  __shared__ float aggS[16][68];   // 68-stride: kills 64-bank conflicts
  __shared__ float ldsS[HIDF];
  __shared__ float ldsQ[HIDF];
  const int tid = threadIdx.x;
  const int l   = tid & 31;
  const int wv  = tid >> 5;
  const int rowBase = blockIdx.x * 16;

  if (tid < HIDF) { ldsS[tid] = 0.f; ldsQ[tid] = 0.f; }

  // ---- build 16 rows of agg1 in LDS: masked temporal mean over 9 children
  for (int rr = 0; rr < 2; ++rr) {
    const int r = wv + rr * 8;
    const int m = rowBase + r;
    const int n = m / 9;
    const int k = m - n * 9;
    int ci; float cts;
    if (k < DD) { ci = nidx[n * DD + k]; cts = nts[n * DD + k]; }
    else        { ci = n;                cts = node_ts[n]; }
    float a0 = 0.f, a1 = 0.f, cnt = 0.f;
#pragma unroll
    for (int j = 0; j < DD + 1; ++j) {
      int gc; bool ok;
      if (j < DD) { gc = nidx[ci * DD + j]; ok = (nts[ci * DD + j] <= cts); }
      else        { gc = ci;                ok = true; }
      if (ok) {
        a0 += x[(size_t)gc * INF + l];
        a1 += x[(size_t)gc * INF + l + 32];
        cnt += 1.f;
      }
    }
    const float inv = 1.f / cnt;
    aggS[r][l]      = a0 * inv;
    aggS[r][l + 32] = a1 * inv;
  }
  __syncthreads();

  // ---- 16x16 tile per wave, K=64 as 16 steps of V_WMMA_F32_16X16X4_F32
  const int row  = l & 15;
  const int koff = (l < 16) ? 0 : 2;
  const int col  = wv * 16 + row;
  v8f c = {};
#pragma unroll
  for (int kk = 0; kk < INF / 4; ++kk) {
    const int kb = kk * 4;
    v2f a, b;
    a.x = aggS[row][kb + koff];
    a.y = aggS[row][kb + koff + 1];
    b.x = W1[(kb + koff)     * HIDF + col];
    b.y = W1[(kb + koff + 1) * HIDF + col];
    c = __builtin_amdgcn_wmma_f32_16x16x4_f32(false, a, false, b, (short)0, c,
                                              false, false);
  }

  // ---- epilogue: bias, store z1, fused column batch-stats
  const float bias = b1[col];
  const int mr0 = (l < 16) ? 0 : 8;
  float s = 0.f, q = 0.f;
#pragma unroll
  for (int v = 0; v < 8; ++v) {
    const float val = c[v] + bias;
    z1[(size_t)(rowBase + mr0 + v) * HIDF + col] = val;
    s += val;
    q += val * val;
  }
  atomicAdd(&ldsS[col], s);
  atomicAdd(&ldsQ[col], q);
  __syncthreads();
  if (tid < HIDF) {
    atomicAdd(&sum1[tid],   ldsS[tid]);
    atomicAdd(&sumsq1[tid], ldsQ[tid]);
  }
}

// ---------------------------------------------------------------------------
// BN stat finalize: mean / invstd per column (biased variance, eps=1e-5)
// ---------------------------------------------------------------------------
__global__ void k_finalize(const float* __restrict__ sum,
                           const float* __restrict__ sumsq,
                           float* __restrict__ mean,
                           float* __restrict__ invstd,
                           int ncols, float invM) {
  int cIdx = threadIdx.x;
  if (cIdx < ncols) {
    float m = sum[cIdx] * invM;
    float v = sumsq[cIdx] * invM - m * m;
    mean[cIdx]   = m;
    invstd[cIdx] = rsqrtf(v + EPSF);
  }
}

// ---------------------------------------------------------------------------
// BN1 + ReLU applied on-the-fly to z1, then masked temporal mean over the
// 9 level-2 children -> agg2[N,128]. One block per node, 128 threads = cols.
// ---------------------------------------------------------------------------
__global__ __launch_bounds__(128)
void k_bn1_agg2(const float* __restrict__ z1,
                const float* __restrict__ node_ts,
                const float* __restrict__ nts,
                const float* __restrict__ mean1,
                const float* __restrict__ invstd1,
                const float* __restrict__ g1,
                const float* __restrict__ be1,
                float* __restrict__ agg2) {
  const int n   = blockIdx.x;
  const int col = threadIdx.x;
  const float nt = node_ts[n];
  const float m1 = mean1[col], is1 = invstd1[col];
  const float gg = g1[col],    bb  = be1[col];
  float acc = 0.f, cnt = 0.f;
#pragma unroll
  for (int k = 0; k < DD + 1; ++k) {
    const bool ok = (k < DD) ? (nts[n * DD + k] <= nt) : true;
    const float h = z1[((size_t)n * 9 + k) * HIDF + col];
    const float hn = fmaxf((h - m1) * is1 * gg + bb, 0.f);
    if (ok) { acc += hn; cnt += 1.f; }
  }
  agg2[(size_t)n * HIDF + col] = acc / cnt;
}

// ---------------------------------------------------------------------------
// z2 = agg2 @ W2 + b2 via WMMA f32 (K=128), fused column batch stats.
// One block = 16 rows x 64 cols (4 waves).
// ---------------------------------------------------------------------------
__global__ __launch_bounds__(128)
void k_gemm2(const float* __restrict__ agg2,
             const float* __restrict__ W2,
             const float* __restrict__ b2,
             float* __restrict__ z2,
             float* __restrict__ sum2,
             float* __restrict__ sumsq2) {
  __shared__ float ldsS[OUTF];
  __shared__ float ldsQ[OUTF];
  const int tid = threadIdx.x;
  const int l   = tid & 31;
  const int wv  = tid >> 5;
  const int rowBase = blockIdx.x * 16;
  if (tid < OUTF) { ldsS[tid] = 0.f; ldsQ[tid] = 0.f; }
  __syncthreads();

  const int row  = l & 15;
  const int koff = (l < 16) ? 0 : 2;
  const int col  = wv * 16 + row;
  v8f c = {};
#pragma unroll
  for (int kk = 0; kk < HIDF / 4; ++kk) {
    const int kb = kk * 4;
    v2f a, b;
    a.x = agg2[(size_t)(rowBase + row) * HIDF + kb + koff];
    a.y = agg2[(size_t)(rowBase + row) * HIDF + kb + koff + 1];
    b.x = W2[(kb + koff)     * OUTF + col];
    b.y = W2[(kb + koff + 1) * OUTF + col];
    c = __builtin_amdgcn_wmma_f32_16x16x4_f32(false, a, false, b, (short)0, c,
                                              false, false);
  }

  const float bias = b2[col];
  const int mr0 = (l < 16) ? 0 : 8;
  float s = 0.f, q = 0.f;
#pragma unroll
  for (int v = 0; v < 8; ++v) {
    const float val = c[v] + bias;
    z2[(size_t)(rowBase + mr0 + v) * OUTF + col] = val;
    s += val;
    q += val * val;
  }
  atomicAdd(&ldsS[col], s);
  atomicAdd(&ldsQ[col], q);
  __syncthreads();
  if (tid < OUTF) {
    atomicAdd(&sum2[tid],   ldsS[tid]);
    atomicAdd(&sumsq2[tid], ldsQ[tid]);
  }
}

// ---------------------------------------------------------------------------
// BN2 + ReLU elementwise, in place on z2 -> embs
// ---------------------------------------------------------------------------
__global__ __launch_bounds__(256)
void k_bn2relu(float* __restrict__ z2,
               const float* __restrict__ mean2,
               const float* __restrict__ invstd2,
               const float* __restrict__ g2,
               const float* __restrict__ be2) {
  const int i = blockIdx.x * 256 + threadIdx.x;   // i < NN*OUTF
  const int col = i & (OUTF - 1);
  const float v = z2[i];
  z2[i] = fmaxf((v - mean2[col]) * invstd2[col] * g2[col] + be2[col], 0.f);
}

// ---------------------------------------------------------------------------
// HadNet decoder + BCEWithLogits: one wave32 per edge, shuffle reduction,
// one global atomic per block.
// ---------------------------------------------------------------------------
__global__ __launch_bounds__(256)
void k_decode(const float* __restrict__ embs,
              const int* __restrict__ target,
              const int* __restrict__ negp,
              const float* __restrict__ Wd,
              const float* __restrict__ bd,
              float* __restrict__ out) {
  __shared__ float red[8];
  const int l  = threadIdx.x & 31;
  const int wv = threadIdx.x >> 5;
  const long e = (long)blockIdx.x * 8 + wv;   // e in [0, 2E)
  int a, b; float label;
  if (e < EE) { a = target[e];       b = target[EE + e];       label = 1.f; }
  else        { a = negp[e - EE];    b = negp[EE + (e - EE)];  label = 0.f; }
  const float* ea = embs + (size_t)a * OUTF;
  const float* eb = embs + (size_t)b * OUTF;
  float p = ea[2 * l]     * eb[2 * l]     * Wd[2 * l]
          + ea[2 * l + 1] * eb[2 * l + 1] * Wd[2 * l + 1];
#pragma unroll
  for (int off = 16; off > 0; off >>= 1) p += __shfl_xor(p, off, 32);
  if (l == 0) {
    p += bd[0];
    red[wv] = fmaxf(p, 0.f) - p * label + log1pf(expf(-fabsf(p)));
  }
  __syncthreads();
  if (threadIdx.x == 0) {
    float s = 0.f;
#pragma unroll
    for (int i = 0; i < 8; ++i) s += red[i];
    atomicAdd(out, s * (0.5f / (float)EE));
  }
}

// ---------------------------------------------------------------------------
extern "C" void kernel_launch(void* const* d_in, const int* in_sizes, int n_in,
                              void* d_out, int out_size, void* d_ws, size_t ws_size,
                              hipStream_t stream) {
  const float* x       = (const float*)d_in[0];
  const float* node_ts = (const float*)d_in[1];
  const int*   nidx    = (const int*)d_in[2];
  const float* nts     = (const float*)d_in[3];
  const int*   target  = (const int*)d_in[4];
  const int*   negp    = (const int*)d_in[5];
  const float* W1  = (const float*)d_in[6];
  const float* b1  = (const float*)d_in[7];
  const float* g1  = (const float*)d_in[8];
  const float* be1 = (const float*)d_in[9];
  const float* W2  = (const float*)d_in[10];
  const float* b2  = (const float*)d_in[11];
  const float* g2  = (const float*)d_in[12];
  const float* be2 = (const float*)d_in[13];
  const float* Wd  = (const float*)d_in[14];
  const float* bd  = (const float*)d_in[15];
  float* out = (float*)d_out;

  float* ws   = (float*)d_ws;
  float* z1   = ws;                                // MM*HIDF  (92 MB)
  float* agg2 = z1   + (size_t)MM * HIDF;          // NN*HIDF
  float* z2   = agg2 + (size_t)NN * HIDF;          // NN*OUTF  (becomes embs)
  float* st   = z2   + (size_t)NN * OUTF;          // 768 floats of stats
  float* sum1    = st;        float* sumsq1 = st + 128;
  float* mean1   = st + 256;  float* invstd1 = st + 384;
  float* sum2    = st + 512;  float* sumsq2 = st + 576;
  float* mean2   = st + 640;  float* invstd2 = st + 704;

  k_init    <<<3, 256, 0, stream>>>(st, out);
  k_gemm1   <<<MM / 16, 256, 0, stream>>>(x, node_ts, nidx, nts, W1, b1,
                                          z1, sum1, sumsq1);
  k_finalize<<<1, 128, 0, stream>>>(sum1, sumsq1, mean1, invstd1, HIDF,
                                    1.0f / (float)MM);
  k_bn1_agg2<<<NN, 128, 0, stream>>>(z1, node_ts, nts, mean1, invstd1, g1, be1,
                                     agg2);
  k_gemm2   <<<NN / 16, 128, 0, stream>>>(agg2, W2, b2, z2, sum2, sumsq2);
  k_finalize<<<1, 64, 0, stream>>>(sum2, sumsq2, mean2, invstd2, OUTF,
                                   1.0f / (float)NN);
  k_bn2relu <<<(NN * OUTF) / 256, 256, 0, stream>>>(z2, mean2, invstd2, g2, be2);
  k_decode  <<<(2 * EE) / 8, 256, 0, stream>>>(z2, target, negp, Wd, bd, out);
}